// ShiftTransportAttention_13460427505681
// MI455X (gfx1250) — compile-verified
//
#include <hip/hip_runtime.h>
#include <hip/hip_bf16.h>
#include <math.h>

// ---------------- problem constants ----------------
#define NB 32
#define NH 128
#define NW 128
#define NC 10
#define MS 5
#define NSHIFT 120          // (2*MS+1)^2 - 1
#define SPAD 128            // shifts padded to 8 tiles of 16
#define BANDS 8
#define BROWS 16            // rows per band (NH / BANDS)
#define PROWS (BROWS + 2*MS)   // 26 padded rows staged per band
#define PAD 8               // horizontal halo pad (>=MS; 8 keeps async B128 16B-aligned)
#define PCOLSP (NW + 2*PAD) // 144 padded cols (row stride)

// K1 dynamic-LDS layout (byte offsets)
#define DIN_BYTES (PROWS * PCOLSP * 4)      // 14976
#define DW_OFF    DIN_BYTES                 // 14976 (16B aligned)
#define DW_BYTES  (BROWS * NW * 8)          // 16384: interleaved {color, weight}
#define WEXP_OFF  (DW_OFF + DW_BYTES)       // 31360
#define K1_LDS    (WEXP_OFF + 64)           // 31424
// K3 dynamic-LDS layout
#define AT_OFF    DIN_BYTES                 // 14976
#define K3_LDS    (AT_OFF + SPAD * 4)       // 15488

typedef __attribute__((ext_vector_type(2))) float v2f;
typedef __attribute__((ext_vector_type(8))) float v8f;

// ---------------- CDNA5 async global->LDS helpers ----------------
// GVS mode: mem_addr = SADDR(u64) + VADDR(i32) + IOFFSET ; VDST = LDS byte addr.
__device__ __forceinline__ void async_b128(unsigned lds_off, const void* base,
                                           unsigned byte_off) {
    asm volatile("global_load_async_to_lds_b128 %0, %1, %2"
                 :: "v"(lds_off), "v"(byte_off), "s"(base) : "memory");
}
__device__ __forceinline__ void async_b32(unsigned lds_off, const void* base,
                                          unsigned byte_off) {
    asm volatile("global_load_async_to_lds_b32 %0, %1, %2"
                 :: "v"(lds_off), "v"(byte_off), "s"(base) : "memory");
}
__device__ __forceinline__ void wait_async0() {
    asm volatile("s_wait_asynccnt 0" ::: "memory");
}

// shift index -> (dy,dx), dy-major, dx inner, skipping (0,0) at flat index 60
__device__ __forceinline__ void shift_of(int s, int& dy, int& dx) {
    int idx = s + (s >= 60 ? 1 : 0);
    dy = idx / 11 - MS;
    dx = idx % 11 - MS;
}

// ---------------- K0: zero the match scratch ----------------
__global__ void k_zero(float* __restrict__ p, int n) {
    int i = blockIdx.x * blockDim.x + threadIdx.x;
    if (i < n) p[i] = 0.0f;
}

// ---------------- K1: match[b][s], async-staged LDS + WMMA reduction -------
// 256 WGs = (b*8 + band); 256 threads = 8 waves; wave w owns shift tile w.
// Lane L accumulates shift row L&15 into the two f32 K-slots given by the
// 16x16x4 f32 A-matrix layout (lanes 0-15 -> K0/K1, lanes 16-31 -> K2/K3);
// one V_WMMA_F32_16X16X4_F32 with B=ones, C=0 reduces the 4 partials/row.
__global__ __launch_bounds__(256) void k_match(
        const int* __restrict__ demo_in, const int* __restrict__ demo_out,
        const float* __restrict__ clw, float* __restrict__ match) {
    extern __shared__ char smem[];
    int*        din  = (int*)smem;                       // [PROWS*PCOLSP]
    int*        dwi  = (int*)(smem + DW_OFF);            // interleaved {c,w}
    const int4* dwq  = (const int4*)(smem + DW_OFF);     // pair view (16B)
    float*      wexp = (float*)(smem + WEXP_OFF);

    const int b    = blockIdx.x >> 3;
    const int band = blockIdx.x & 7;
    const int tid  = threadIdx.x;
    const int row0 = band * BROWS;

    if (tid < 16) wexp[tid] = (tid < NC) ? expf(clw[tid]) : 0.0f;

    // Zero only the halo cells (disjoint from async targets -> no ordering hazard)
    for (int i = tid; i < PROWS * PCOLSP; i += 256) {
        int r  = i / PCOLSP, cix = i % PCOLSP;
        int gy = row0 - MS + r, gx = cix - PAD;
        if (gy < 0 || gy >= NH || gx < 0 || gx >= NW) din[i] = 0;
    }

    // Async-stage in-bounds demo_in rows: 128 ints per row as 32 x B128
    for (int i = tid; i < PROWS * 32; i += 256) {
        int r = i >> 5, seg = i & 31;
        int gy = row0 - MS + r;
        if (gy >= 0 && gy < NH) {
            unsigned lds_off = (unsigned)((r * PCOLSP + PAD + seg * 4) * 4);
            unsigned gb      = (unsigned)(((b * NH + gy) * NW + seg * 4) * 4);
            async_b128(lds_off, demo_in, gb);
        }
    }
    // Async-scatter demo_out colors into dw[2p] (8-byte stride)
    {
        unsigned gbase = (unsigned)(((b * NH + row0) * NW) * 4);
        for (int i = tid; i < BROWS * NW; i += 256)
            async_b32((unsigned)(DW_OFF + i * 8), demo_out, gbase + (unsigned)(i * 4));
    }
    wait_async0();       // each wave drains its own ASYNCcnt ...
    __syncthreads();     // ... then the barrier makes all waves' data visible

    // Fill dw[2p+1] = exp(clw)[color]
    for (int i = tid; i < BROWS * NW; i += 256) {
        int c = dwi[2 * i];
        dwi[2 * i + 1] = __float_as_int(wexp[c]);
    }
    __syncthreads();

    const int wave = tid >> 5;
    const int lane = tid & 31;
    const int m    = lane & 15;     // shift row within tile
    const int hi   = lane >> 4;     // K-slot group
    const int s    = wave * 16 + m;

    int dy = 0, dx = 0;
    if (s < NSHIFT) shift_of(s, dy, dx);

    float a0 = 0.0f, a1 = 0.0f;
    // 2048 band pixels split 4 ways by K-slot; each lane takes 2 adjacent slots.
    // fmaf(mask, w, acc) keeps the weight load unconditional: the select is
    // between the constants 1.0/0.0, so no load can be sunk into a branch.
    for (int base = 0; base < BROWS * NW; base += 4) {
        int p0 = base + (hi << 1);
        int y0 = p0 >> 7, x0 = p0 & (NW - 1);
        int roff = (y0 + MS - dy) * PCOLSP + (x0 + PAD - dx);
        int v0 = din[roff];
        int v1 = din[roff + 1];
        int4 q = dwq[p0 >> 1];      // {c0,w0,c1,w1}
        float m0 = (v0 == q.x) ? 1.0f : 0.0f;
        float m1 = (v1 == q.z) ? 1.0f : 0.0f;
        a0 = fmaf(m0, __int_as_float(q.y), a0);
        a1 = fmaf(m1, __int_as_float(q.w), a1);
    }

    // Cross-lane row reduction via WMMA: D = A(16x4) * ones(4x16) + 0
    v2f a    = { a0, a1 };
    v2f ones = { 1.0f, 1.0f };
    v8f cz   = {};
    v8f d = __builtin_amdgcn_wmma_f32_16x16x4_f32(
        /*neg_a=*/false, a, /*neg_b=*/false, ones,
        /*c_mod=*/(short)0, cz, /*reuse_a=*/false, /*reuse_b=*/false);

    // Rows 0-7 live in lane 0 (col 0); rows 8-15 in lane 16. Only wave 7's
    // upper half (s=120..127) is padding -> wave-uniform guard, no per-element
    // exec juggling.
    if (lane == 0) {
        #pragma unroll
        for (int j = 0; j < 8; ++j)
            atomicAdd(&match[b * SPAD + wave * 16 + j], d[j]);
    } else if (lane == 16 && wave < 7) {
        #pragma unroll
        for (int j = 0; j < 8; ++j)
            atomicAdd(&match[b * SPAD + wave * 16 + 8 + j], d[j]);
    }
}

// ---------------- K2: per-batch softmax over 120 shifts ----------------
__global__ __launch_bounds__(128) void k_softmax(
        const float* __restrict__ match, const float* __restrict__ lscale,
        float* __restrict__ attn) {
    __shared__ float red[128];
    const int b = blockIdx.x;
    const int t = threadIdx.x;

    const float scale = fminf(expf(lscale[0]), 100.0f) * (1.0f / 128.0f); // /sqrt(H*W)
    float sc = (t < NSHIFT) ? match[b * SPAD + t] * scale : -3.4e38f;

    red[t] = sc;
    __syncthreads();
    for (int off = 64; off > 0; off >>= 1) {
        if (t < off) red[t] = fmaxf(red[t], red[t + off]);
        __syncthreads();
    }
    const float mx = red[0];
    __syncthreads();

    float e = (t < NSHIFT) ? expf(sc - mx) : 0.0f;
    red[t] = e;
    __syncthreads();
    for (int off = 64; off > 0; off >>= 1) {
        if (t < off) red[t] += red[t + off];
        __syncthreads();
    }
    attn[b * SPAD + t] = e * (1.0f / red[0]);
}

// ---------------- K3: attention-weighted color gather + log ----------------
// fmaf-mask form: select is between constants only -> branch-free cndmask+fmac.
#define ACC10(c_, a_)                                      \
    A0 = fmaf(((c_) == 0) ? 1.0f : 0.0f, (a_), A0);        \
    A1 = fmaf(((c_) == 1) ? 1.0f : 0.0f, (a_), A1);        \
    A2 = fmaf(((c_) == 2) ? 1.0f : 0.0f, (a_), A2);        \
    A3 = fmaf(((c_) == 3) ? 1.0f : 0.0f, (a_), A3);        \
    A4 = fmaf(((c_) == 4) ? 1.0f : 0.0f, (a_), A4);        \
    A5 = fmaf(((c_) == 5) ? 1.0f : 0.0f, (a_), A5);        \
    A6 = fmaf(((c_) == 6) ? 1.0f : 0.0f, (a_), A6);        \
    A7 = fmaf(((c_) == 7) ? 1.0f : 0.0f, (a_), A7);        \
    A8 = fmaf(((c_) == 8) ? 1.0f : 0.0f, (a_), A8);        \
    A9 = fmaf(((c_) == 9) ? 1.0f : 0.0f, (a_), A9);

__global__ __launch_bounds__(256) void k_out(
        const int* __restrict__ query_in, const float* __restrict__ attn,
        float* __restrict__ out) {
    extern __shared__ char smem[];
    int*   qin = (int*)smem;                  // [PROWS*PCOLSP]
    float* at  = (float*)(smem + AT_OFF);     // [SPAD]

    const int b    = blockIdx.x >> 3;
    const int band = blockIdx.x & 7;
    const int tid  = threadIdx.x;
    const int row0 = band * BROWS;

    // Zero halo cells
    for (int i = tid; i < PROWS * PCOLSP; i += 256) {
        int r  = i / PCOLSP, cix = i % PCOLSP;
        int gy = row0 - MS + r, gx = cix - PAD;
        if (gy < 0 || gy >= NH || gx < 0 || gx >= NW) qin[i] = 0;
    }
    // Async-stage in-bounds query rows
    for (int i = tid; i < PROWS * 32; i += 256) {
        int r = i >> 5, seg = i & 31;
        int gy = row0 - MS + r;
        if (gy >= 0 && gy < NH) {
            unsigned lds_off = (unsigned)((r * PCOLSP + PAD + seg * 4) * 4);
            unsigned gb      = (unsigned)(((b * NH + gy) * NW + seg * 4) * 4);
            async_b128(lds_off, query_in, gb);
        }
    }
    if (tid < SPAD) at[tid] = attn[b * SPAD + tid];
    wait_async0();
    __syncthreads();

    for (int k = 0; k < 8; ++k) {
        const int p = tid + (k << 8);
        const int y = p >> 7, x = p & (NW - 1);

        float A0 = 0, A1 = 0, A2 = 0, A3 = 0, A4 = 0;
        float A5 = 0, A6 = 0, A7 = 0, A8 = 0, A9 = 0;

        int s = 0;
        for (int dy = -MS; dy <= MS; ++dy) {
            const int rbase = (y + MS - dy) * PCOLSP + (x + PAD);
            #pragma unroll
            for (int dx = -MS; dx <= MS; ++dx) {
                if (dy == 0 && dx == 0) continue;
                const float a = at[s++];
                const int   c = qin[rbase - dx];
                ACC10(c, a)
            }
        }

        const int gy = row0 + y;
        const size_t obase = ((size_t)b * NC) * (NH * NW) + (size_t)gy * NW + x;
        out[obase + 0 * (NH * NW)] = logf(fmaxf(A0, 1e-6f));
        out[obase + 1 * (NH * NW)] = logf(fmaxf(A1, 1e-6f));
        out[obase + 2 * (NH * NW)] = logf(fmaxf(A2, 1e-6f));
        out[obase + 3 * (NH * NW)] = logf(fmaxf(A3, 1e-6f));
        out[obase + 4 * (NH * NW)] = logf(fmaxf(A4, 1e-6f));
        out[obase + 5 * (NH * NW)] = logf(fmaxf(A5, 1e-6f));
        out[obase + 6 * (NH * NW)] = logf(fmaxf(A6, 1e-6f));
        out[obase + 7 * (NH * NW)] = logf(fmaxf(A7, 1e-6f));
        out[obase + 8 * (NH * NW)] = logf(fmaxf(A8, 1e-6f));
        out[obase + 9 * (NH * NW)] = logf(fmaxf(A9, 1e-6f));
    }
}

// ---------------- launch ----------------
extern "C" void kernel_launch(void* const* d_in, const int* in_sizes, int n_in,
                              void* d_out, int out_size, void* d_ws, size_t ws_size,
                              hipStream_t stream) {
    const int*   demo_in  = (const int*)d_in[0];
    const int*   demo_out = (const int*)d_in[1];
    const int*   query_in = (const int*)d_in[2];
    const float* clw      = (const float*)d_in[3];
    const float* lscale   = (const float*)d_in[4];
    float*       out      = (float*)d_out;

    float* match = (float*)d_ws;        // [NB][SPAD]
    float* attn  = match + NB * SPAD;   // [NB][SPAD]

    k_zero<<<(NB * SPAD + 255) / 256, 256, 0, stream>>>(match, NB * SPAD);
    k_match<<<NB * BANDS, 256, K1_LDS, stream>>>(demo_in, demo_out, clw, match);
    k_softmax<<<NB, 128, 0, stream>>>(match, lscale, attn);
    k_out<<<NB * BANDS, 256, K3_LDS, stream>>>(query_in, attn, out);
}